// AttnDecoderCell_74131135529118
// MI455X (gfx1250) — compile-verified
//
#include <hip/hip_runtime.h>
#include <math.h>

#define HDIM  256
#define TDIM  1024
#define VOCAB 32000
#define TCHUNKS 4            // attention T-dim split for occupancy
#define TCHUNK  (TDIM / TCHUNKS)

typedef __attribute__((ext_vector_type(2))) float v2f;
typedef __attribute__((ext_vector_type(4))) float v4f;
typedef __attribute__((ext_vector_type(8))) float v8f;

// gfx1250 hardware transcendental tanh (v_tanh_f32); fallback is a cheap
// v_exp_f32-based form (correctly saturates to +/-1), never the slow ocml tanh.
__device__ __forceinline__ float fast_tanh(float x) {
#if __has_builtin(__builtin_amdgcn_tanhf)
  return __builtin_amdgcn_tanhf(x);
#elif __has_builtin(__builtin_amdgcn_tanh_f32)
  return __builtin_amdgcn_tanh_f32(x);
#else
  float t = __expf(2.0f * x);
  return 1.0f - 2.0f / (t + 1.0f);
#endif
}

// ---------------------------------------------------------------------------
// Kernel 1a: attention partials. Grid = 256 columns x 4 T-chunks (j-major so
// consecutive blocks share W/U columns in L2). Block = 256 threads, one t per
// thread. Produces per-(j,chunk) online-softmax stats:
//   Mp = max_t e[t,j], Sp = sum exp(e-Mp), Cp = sum exp(e-Mp)*h[t,j]
// ---------------------------------------------------------------------------
__global__ void attn_partial_kernel(const float* __restrict__ g,
                                    const float* __restrict__ h,
                                    const float* __restrict__ W,
                                    const float* __restrict__ U,
                                    const float* __restrict__ v,
                                    float* __restrict__ M_part,
                                    float* __restrict__ S_part,
                                    float* __restrict__ C_part) {
  const int j     = blockIdx.x >> 2;
  const int chunk = blockIdx.x & 3;
  const int tid   = threadIdx.x;
  __shared__ float sA[HDIM];   // W[i,j]*g[j]
  __shared__ float sB[HDIM];   // U[i,j]
  __shared__ float sV[HDIM];   // v[i]
  __shared__ float red0[256];
  __shared__ float red1[256];

  const float gj = g[j];
  sA[tid] = W[tid * HDIM + j] * gj;
  sB[tid] = U[tid * HDIM + j];
  sV[tid] = v[tid];
  __syncthreads();

  const int t = chunk * TCHUNK + tid;
  const float y = h[(size_t)t * HDIM + j];
  float e = 0.0f;

  for (int i = 0; i < HDIM; i += 4) {
    v4f a4 = *(const v4f*)&sA[i];
    v4f b4 = *(const v4f*)&sB[i];
    v4f w4 = *(const v4f*)&sV[i];
#pragma unroll
    for (int u = 0; u < 4; ++u) {
      e += w4[u] * fast_tanh(fmaf(b4[u], y, a4[u]));
    }
  }

  // block max
  red0[tid] = e;
  __syncthreads();
  for (int off = 128; off > 0; off >>= 1) {
    if (tid < off) red0[tid] = fmaxf(red0[tid], red0[tid + off]);
    __syncthreads();
  }
  const float M = red0[0];
  __syncthreads();

  // block sum of exp and weighted sum
  const float p = __expf(e - M);
  red0[tid] = p;
  red1[tid] = p * y;
  __syncthreads();
  for (int off = 128; off > 0; off >>= 1) {
    if (tid < off) {
      red0[tid] += red0[tid + off];
      red1[tid] += red1[tid + off];
    }
    __syncthreads();
  }
  if (tid == 0) {
    M_part[chunk * HDIM + j] = M;
    S_part[chunk * HDIM + j] = red0[0];
    C_part[chunk * HDIM + j] = red1[0];
  }
}

// ---------------------------------------------------------------------------
// Kernel 1b: combine the 4 per-chunk partials per column -> c[j].
// ---------------------------------------------------------------------------
__global__ void attn_combine_kernel(const float* __restrict__ M_part,
                                    const float* __restrict__ S_part,
                                    const float* __restrict__ C_part,
                                    float* __restrict__ c_out) {
  const int j = threadIdx.x;  // 0..255
  float M = -INFINITY, S = 0.0f, C = 0.0f;
#pragma unroll
  for (int ch = 0; ch < TCHUNKS; ++ch) {
    float m = M_part[ch * HDIM + j];
    float s = S_part[ch * HDIM + j];
    float c = C_part[ch * HDIM + j];
    float newM = fmaxf(M, m);
    float f0 = __expf(M - newM);
    float f1 = __expf(m - newM);
    S = S * f0 + s * f1;
    C = C * f0 + c * f1;
    M = newM;
  }
  c_out[j] = C / S;
}

// ---------------------------------------------------------------------------
// Kernel 2: h1 = tanh(W_ih@x + b_ih + W_hh@[g;c] + b_hh). 512 threads, one
// output each (768-MAC dot). Also emits new_hidden = h1[:256].
// ---------------------------------------------------------------------------
__global__ void rnn_cell_kernel(const int* __restrict__ input,
                                const float* __restrict__ g,
                                const float* __restrict__ c,
                                const float* __restrict__ emb,
                                const float* __restrict__ W_ih,
                                const float* __restrict__ W_hh,
                                const float* __restrict__ b_ih,
                                const float* __restrict__ b_hh,
                                float* __restrict__ h1,
                                float* __restrict__ new_hidden) {
  __shared__ float sx[HDIM];
  __shared__ float sh0[2 * HDIM];
  const int tid = threadIdx.x;  // 0..511
  const int idx = input[0];
  if (tid < HDIM) {
    sx[tid]  = emb[(size_t)idx * HDIM + tid];
    sh0[tid] = g[tid];
  } else {
    sh0[tid] = c[tid - HDIM];
  }
  __syncthreads();

  float acc = b_ih[tid] + b_hh[tid];
  const float* wih = W_ih + (size_t)tid * HDIM;
  for (int k = 0; k < HDIM; k += 4) {
    v4f wv = *(const v4f*)(wih + k);
    v4f xv = *(const v4f*)&sx[k];
#pragma unroll
    for (int u = 0; u < 4; ++u) acc = fmaf(wv[u], xv[u], acc);
  }
  const float* whh = W_hh + (size_t)tid * (2 * HDIM);
  for (int k = 0; k < 2 * HDIM; k += 4) {
    v4f wv = *(const v4f*)(whh + k);
    v4f hv = *(const v4f*)&sh0[k];
#pragma unroll
    for (int u = 0; u < 4; ++u) acc = fmaf(wv[u], hv[u], acc);
  }
  float r = fast_tanh(acc);
  h1[tid] = r;
  if (tid < HDIM) new_hidden[tid] = r;
}

// ---------------------------------------------------------------------------
// Kernel 3: logits = W_out @ h1 + b_out via V_WMMA_F32_16X16X4_F32.
// 250 blocks x 256 threads (8 waves); wave owns a 16-row tile of W_out.
// A (32-bit 16x4, ISA 7.12.2): lanes 0-15 carry K=0/1 for row M=lane; lanes
// 16-31 carry K=2/3 for row M=lane-16. B (4x16): only column 0 populated
// (lane 0 -> K0,K1; lane 16 -> K2,K3), built with selects so EXEC stays
// all-ones across the WMMA loop. Column 0 of D: lane 0 holds M=0..7 in
// acc[0..7], lane 16 holds M=8..15. Bandwidth-bound: 65.5 MB / 23.3 TB/s.
// ---------------------------------------------------------------------------
__global__ void out_gemv_wmma_kernel(const float* __restrict__ W_out,
                                     const float* __restrict__ h1,
                                     const float* __restrict__ b_out,
                                     float* __restrict__ logits) {
  __shared__ float sh1[2 * HDIM];  // 512 floats
  const int tid = threadIdx.x;     // 0..255
  sh1[tid]       = h1[tid];
  sh1[tid + 256] = h1[tid + 256];
  __syncthreads();

  const int wave = tid >> 5;
  const int lane = tid & 31;
  const int tile = blockIdx.x * 8 + wave;   // 0..1999
  const int r0   = tile * 16;
  const int row  = r0 + (lane & 15);
  const int ksub = (lane >> 4) << 1;        // 0 for lanes 0-15, 2 for 16-31
  const bool bact = (lane & 15) == 0;       // lanes 0 and 16 supply B col 0

  const float* wrow = W_out + (size_t)row * (2 * HDIM) + ksub;

  v8f acc = {0.f, 0.f, 0.f, 0.f, 0.f, 0.f, 0.f, 0.f};
  for (int k = 0; k < 2 * HDIM; k += 4) {
    v2f a = *(const v2f*)(wrow + k);                 // global_load_b64
    v2f bv = *(const v2f*)&sh1[k + ksub];            // LDS broadcast
    v2f b;
    b.x = bact ? bv.x : 0.0f;
    b.y = bact ? bv.y : 0.0f;
    acc = __builtin_amdgcn_wmma_f32_16x16x4_f32(
        /*neg_a=*/false, a, /*neg_b=*/false, b,
        /*c_mod=*/(short)0, acc, /*reuse_a=*/false, /*reuse_b=*/false);
  }

  if (lane == 0) {
#pragma unroll
    for (int p = 0; p < 8; ++p) logits[r0 + p] = acc[p] + b_out[r0 + p];
  } else if (lane == 16) {
#pragma unroll
    for (int p = 0; p < 8; ++p) logits[r0 + 8 + p] = acc[p] + b_out[r0 + 8 + p];
  }
}

// ---------------------------------------------------------------------------
// Kernel 4: log_softmax over 32000 logits. One 1024-thread block; each thread
// keeps its 32 strided values in registers between the two reductions.
// ---------------------------------------------------------------------------
__global__ void logsoftmax_kernel(const float* __restrict__ logits,
                                  float* __restrict__ out) {
  __shared__ float red[1024];
  const int tid = threadIdx.x;
  float vals[32];
  float m = -INFINITY;
#pragma unroll
  for (int q = 0; q < 32; ++q) {
    int idx = tid + q * 1024;
    float x = (idx < VOCAB) ? logits[idx] : -INFINITY;
    vals[q] = x;
    m = fmaxf(m, x);
  }
  red[tid] = m;
  __syncthreads();
  for (int off = 512; off > 0; off >>= 1) {
    if (tid < off) red[tid] = fmaxf(red[tid], red[tid + off]);
    __syncthreads();
  }
  const float M = red[0];
  __syncthreads();

  float s = 0.0f;
#pragma unroll
  for (int q = 0; q < 32; ++q) {
    int idx = tid + q * 1024;
    if (idx < VOCAB) s += __expf(vals[q] - M);
  }
  red[tid] = s;
  __syncthreads();
  for (int off = 512; off > 0; off >>= 1) {
    if (tid < off) red[tid] += red[tid + off];
    __syncthreads();
  }
  const float lse = M + logf(red[0]);
#pragma unroll
  for (int q = 0; q < 32; ++q) {
    int idx = tid + q * 1024;
    if (idx < VOCAB) out[idx] = vals[q] - lse;
  }
}

// ---------------------------------------------------------------------------
extern "C" void kernel_launch(void* const* d_in, const int* in_sizes, int n_in,
                              void* d_out, int out_size, void* d_ws, size_t ws_size,
                              hipStream_t stream) {
  (void)in_sizes; (void)n_in; (void)out_size; (void)ws_size;

  const int*   input = (const int*)d_in[0];
  const float* g     = (const float*)d_in[1];   // input_hidden (1,1,256)
  const float* h     = (const float*)d_in[2];   // encoder_hiddens (1024,1,256)
  const float* W     = (const float*)d_in[3];
  const float* U     = (const float*)d_in[4];
  const float* v     = (const float*)d_in[5];
  const float* emb   = (const float*)d_in[6];
  const float* W_ih  = (const float*)d_in[7];
  const float* W_hh  = (const float*)d_in[8];
  const float* b_ih  = (const float*)d_in[9];
  const float* b_hh  = (const float*)d_in[10];
  const float* W_out = (const float*)d_in[11];
  const float* b_out = (const float*)d_in[12];

  float* out = (float*)d_out;        // [0,32000) log-probs, [32000,32256) new hidden
  float* ws  = (float*)d_ws;
  float* c_ctx  = ws;                // 256 floats
  float* h1     = ws + 256;          // 512 floats
  float* logits = ws + 768;          // 32000 floats
  // Attention partials reuse the logits region (attention finishes before the
  // GEMV writes logits): 3 arrays of TCHUNKS*HDIM = 1024 floats each.
  float* M_part = logits;            // 1024
  float* S_part = logits + 1024;     // 1024
  float* C_part = logits + 2048;     // 1024

  attn_partial_kernel<<<HDIM * TCHUNKS, 256, 0, stream>>>(g, h, W, U, v,
                                                          M_part, S_part, C_part);
  attn_combine_kernel<<<1, HDIM, 0, stream>>>(M_part, S_part, C_part, c_ctx);
  rnn_cell_kernel<<<1, 512, 0, stream>>>(input, g, c_ctx, emb, W_ih, W_hh,
                                         b_ih, b_hh, h1, out + VOCAB);
  out_gemv_wmma_kernel<<<VOCAB / 128, 256, 0, stream>>>(W_out, h1, b_out, logits);
  logsoftmax_kernel<<<1, 1024, 0, stream>>>(logits, out);
}